// FcCaps_80238579023896
// MI455X (gfx1250) — compile-verified
//
#include <hip/hip_runtime.h>
#include <math.h>

// Problem constants (from reference setup_inputs)
#define BB  64
#define HH  14
#define WW  14
#define CC  32
#define OO  10
#define OP  12           // padded O (weights zero for o=10,11) -> branch-free B loads
#define SS  (HH*WW)      // 196 spatial positions
#define NN  (SS*CC)      // 6272
#define EPSF 1e-9f

typedef __attribute__((ext_vector_type(2))) float v2f;
typedef __attribute__((ext_vector_type(8))) float v8f;

// ---------------------------------------------------------------------------
// Vote recompute core: one wave covers 4 spatial positions (s0..s0+3), one c,
// all 10 (padded 12) output capsules via 3x V_WMMA_F32_16X16X4_F32.
//   A (16x4): rows m=(p,i), K=j.  lane(m=L&15, hi=L>>4) holds A[m,2hi..2hi+1]
//   B (4x16): cols (q,k), rows K=j. lane col=L&15, VGPR v = B[v+2hi, col]
//   D (16x16): lane col=L&15, VGPR r = D[r+8hi, col]
// Coord-addition offsets ride in the C operand (k==3 columns).
// ap    : lane's pose address for c==0 (advances 16 floats per c)
// wl    : per-lane LDS offset  16*q + k + (2*hi)*4   (o and j folded in below)
// ---------------------------------------------------------------------------
__device__ __forceinline__ void compute_votes(
    const float* __restrict__ ap, const float* __restrict__ w_lds,
    int c, int wl, const v8f& cinit, v8f acc[3])
{
    const float* apc = ap + (size_t)c * 16;
    v2f a;
    a.x = apc[0];
    a.y = apc[1];
    const int wb = c * (OP * 16) + wl;       // + 64*g selects o block
#pragma unroll
    for (int g = 0; g < 3; ++g) {
        v2f bf;
        bf.x = w_lds[wb + 64 * g];           // j = 2*hi
        bf.y = w_lds[wb + 64 * g + 4];       // j = 2*hi + 1
        acc[g] = __builtin_amdgcn_wmma_f32_16x16x4_f32(
            /*neg_a=*/false, a, /*neg_b=*/false, bf,
            /*c_mod=*/(short)0, cinit, /*reuse_a=*/false, /*reuse_b=*/false);
    }
}

// C operand with coord-addition constants:
// d==3 -> (i=0,k=3) += (w+0.5)/14 ;  d==7 -> (i=1,k=3) += (h+0.5)/14
__device__ __forceinline__ v8f make_cinit(int s0, int hi, int k)
{
    v8f c;
#pragma unroll
    for (int r = 0; r < 8; ++r) c[r] = 0.0f;
    if (k == 3) {
#pragma unroll
        for (int r = 0; r < 8; ++r) {
            const int p = 2 * hi + (r >> 2);
            const int s = s0 + p;
            if ((r & 3) == 0)      c[r] = ((float)(s % WW) + 0.5f) / (float)WW;
            else if ((r & 3) == 1) c[r] = ((float)(s / WW) + 0.5f) / (float)HH;
        }
    }
    return c;
}

// Stage weights into LDS, padded to 12 output capsules (zeros beyond 10).
__device__ __forceinline__ void stage_weights(const float* __restrict__ w,
                                              float* __restrict__ w_lds, int tid)
{
    for (int idx = tid; idx < CC * OP * 16; idx += 224) w_lds[idx] = 0.0f;
    __syncthreads();
    for (int idx = tid; idx < CC * OO * 16; idx += 224) {
        const int co = idx >> 4;             // c*OO + o
        const int t  = idx & 15;
        const int c  = co / OO, o = co - c * OO;
        w_lds[(c * OP + o) * 16 + t] = w[idx];
    }
    __syncthreads();
}

// ---------------------------------------------------------------------------
// Pass 1: weighted moments  S0[o]=Σ rr_p, S1[o,d]=Σ rr_p*v, S2[o,d]=Σ rr_p*v²
// grid (7, B), block 224 (7 waves, each wave = 4 spatial positions)
// ---------------------------------------------------------------------------
template <int INIT>
__global__ void __launch_bounds__(224)
moments_kernel(const float* __restrict__ pose, const float* __restrict__ act,
               const float* __restrict__ w,    const float* __restrict__ rr,
               float* __restrict__ S0, float* __restrict__ S1, float* __restrict__ S2)
{
    __shared__ float w_lds[CC * OP * 16];
    const int tid = threadIdx.x;
    stage_weights(w, w_lds, tid);

    const int b    = blockIdx.y;
    const int wave = tid >> 5;
    const int L    = tid & 31;
    const int s0   = (blockIdx.x * 7 + wave) * 4;
    const int hi   = L >> 4;
    const int mrow = L & 15;
    const int q    = mrow >> 2;
    const int k    = mrow & 3;
    const int wl   = 16 * q + k + (2 * hi) * 4;

    const v8f cinit = make_cinit(s0, hi, k);

    // hoisted lane base pointers (advance by fixed strides over c)
    const int p = mrow >> 2, i = mrow & 3;
    const float* ap   = pose + ((size_t)b * NN + (s0 + p) * CC) * 16 + i * 4 + 2 * hi;
    const int p_lo = 2 * hi, p_hi = 2 * hi + 1;
    const float* actl = act + (size_t)b * NN + (s0 + p_lo) * CC;
    const float* acth = act + (size_t)b * NN + (s0 + p_hi) * CC;
    const float* rrl[3];
    const float* rrh[3];
    bool  valid[3];
#pragma unroll
    for (int g = 0; g < 3; ++g) {
        const int o  = 4 * g + q;
        valid[g]     = (o < OO);
        const int oc = valid[g] ? o : (OO - 1);
        rrl[g] = rr + ((size_t)b * NN + (s0 + p_lo) * CC) * OO + oc;
        rrh[g] = rr + ((size_t)b * NN + (s0 + p_hi) * CC) * OO + oc;
    }

    float aS0[3]    = {0.f, 0.f, 0.f};
    float aS1[3][4] = {};
    float aS2[3][4] = {};

#pragma unroll 2
    for (int c = 0; c < CC; ++c) {
        v8f acc[3];
        compute_votes(ap, w_lds, c, wl, cinit, acc);

        const float alo = actl[c];
        const float ahi = acth[c];

#pragma unroll
        for (int g = 0; g < 3; ++g) {
            float wlo, whi;
            if (INIT) {               // iteration 0: rr = 1/O uniformly
                wlo = alo * (1.0f / OO);
                whi = ahi * (1.0f / OO);
            } else {
                const float rl = rrl[g][(size_t)c * OO];   // clamped addr, mask value
                const float rh = rrh[g][(size_t)c * OO];
                wlo = valid[g] ? rl * alo : 0.0f;
                whi = valid[g] ? rh * ahi : 0.0f;
            }
            if (k == 0) aS0[g] += wlo + whi;
#pragma unroll
            for (int i2 = 0; i2 < 4; ++i2) {
                const float v0 = acc[g][i2], v1 = acc[g][4 + i2];
                aS1[g][i2] += wlo * v0      + whi * v1;
                aS2[g][i2] += wlo * v0 * v0 + whi * v1 * v1;
            }
        }
    }

    // reduce across hi halves (same (o,k), different p pair), then atomics
#pragma unroll
    for (int g = 0; g < 3; ++g) {
        const int o = 4 * g + q;
        const float r0 = aS0[g] + __shfl_xor(aS0[g], 16, 32);
        float r1[4], r2[4];
#pragma unroll
        for (int i2 = 0; i2 < 4; ++i2) {
            r1[i2] = aS1[g][i2] + __shfl_xor(aS1[g][i2], 16, 32);
            r2[i2] = aS2[g][i2] + __shfl_xor(aS2[g][i2], 16, 32);
        }
        if (hi == 0 && o < OO) {
            if (k == 0) atomicAdd(&S0[b * OO + o], r0);
#pragma unroll
            for (int i2 = 0; i2 < 4; ++i2) {
                atomicAdd(&S1[((size_t)b * OO + o) * 16 + i2 * 4 + k], r1[i2]);
                atomicAdd(&S2[((size_t)b * OO + o) * 16 + i2 * 4 + k], r2[i2]);
            }
        }
    }
}

// ---------------------------------------------------------------------------
// Pass 2: per (b,o) statistics -> mu, var, act_out(sigmoid), log(act+eps)
// grid B, block 160 (= O*16, thread t -> o=t/16, d=t%16)
// ---------------------------------------------------------------------------
__global__ void __launch_bounds__(160)
stats_kernel(const float* __restrict__ S0, const float* __restrict__ S1,
             const float* __restrict__ S2,
             const float* __restrict__ beta_a, const float* __restrict__ beta_u,
             float* __restrict__ mu, float* __restrict__ var_,
             float* __restrict__ logact,
             float inv_temp, int final_,
             float* __restrict__ out_pose, float* __restrict__ out_act)
{
    const int b = blockIdx.x;
    const int t = threadIdx.x;
    const int o = t >> 4, d = t & 15;

    const float rs = S0[b * OO + o] + EPSF;
    const float m  = S1[((size_t)b * OO + o) * 16 + d] / rs;
    const float e2 = S2[((size_t)b * OO + o) * 16 + d] / rs;
    float v = e2 - m * m + EPSF;
    v = fmaxf(v, EPSF);

    mu  [((size_t)b * OO + o) * 16 + d] = m;
    var_[((size_t)b * OO + o) * 16 + d] = v;

    float cost = (beta_u[o] + 0.5f * logf(v)) * rs;
    cost += __shfl_xor(cost, 1, 16);
    cost += __shfl_xor(cost, 2, 16);
    cost += __shfl_xor(cost, 4, 16);
    cost += __shfl_xor(cost, 8, 16);

    if (d == 0) {
        const float a = 1.0f / (1.0f + expf(-inv_temp * (beta_a[o] - cost)));
        logact[b * OO + o] = logf(a + EPSF);
        if (final_) out_act[b * OO + o] = a;
    }
    if (final_) out_pose[((size_t)b * OO + o) * 16 + d] = m;
}

// ---------------------------------------------------------------------------
// Pass 3: rr update. Recompute votes via WMMA, ln_p = -0.5 Σ_d ((v-mu)²/var
// + log(2π var)), softmax over o, write rr. Same grid shape as moments.
// ---------------------------------------------------------------------------
__global__ void __launch_bounds__(224)
rrupd_kernel(const float* __restrict__ pose, const float* __restrict__ w,
             const float* __restrict__ mu, const float* __restrict__ var_,
             const float* __restrict__ logact, float* __restrict__ rr)
{
    __shared__ float w_lds[CC * OP * 16];
    const int tid = threadIdx.x;
    stage_weights(w, w_lds, tid);

    const int b    = blockIdx.y;
    const int wave = tid >> 5;
    const int L    = tid & 31;
    const int s0   = (blockIdx.x * 7 + wave) * 4;
    const int hi   = L >> 4;
    const int mrow = L & 15;
    const int q    = mrow >> 2;
    const int k    = mrow & 3;
    const int wl   = 16 * q + k + (2 * hi) * 4;

    const v8f cinit = make_cinit(s0, hi, k);

    const int p = mrow >> 2, i = mrow & 3;
    const float* ap = pose + ((size_t)b * NN + (s0 + p) * CC) * 16 + i * 4 + 2 * hi;

    // per-lane stats for its (o, k) column, all i
    float mul[3][4], iva[3][4], lcl[3][4], la[3];
#pragma unroll
    for (int g = 0; g < 3; ++g) {
        const int o      = 4 * g + q;
        const bool valid = (o < OO);
        const int  oc    = valid ? o : (OO - 1);
        la[g] = valid ? logact[b * OO + oc] : -1e30f;
#pragma unroll
        for (int i2 = 0; i2 < 4; ++i2) {
            const float mm = mu  [((size_t)b * OO + oc) * 16 + i2 * 4 + k];
            const float vv = var_[((size_t)b * OO + oc) * 16 + i2 * 4 + k];
            mul[g][i2] = valid ? mm : 0.0f;
            iva[g][i2] = valid ? 1.0f / vv : 0.0f;
            lcl[g][i2] = valid ? logf(6.2831853071795864f * vv) : 0.0f;
        }
    }

    for (int c = 0; c < CC; ++c) {
        v8f acc[3];
        compute_votes(ap, w_lds, c, wl, cinit, acc);

        float rrv[2][3];
#pragma unroll
        for (int slot = 0; slot < 2; ++slot) {    // p = 2*hi + slot
            float tg[3];
#pragma unroll
            for (int g = 0; g < 3; ++g) {
                float t = 0.0f;
#pragma unroll
                for (int i2 = 0; i2 < 4; ++i2) {
                    const float dv = acc[g][slot * 4 + i2] - mul[g][i2];
                    t += dv * dv * iva[g][i2] + lcl[g][i2];
                }
                tg[g] = t;
            }
            // sum over k (4 adjacent lanes share (o, p-pair))
#pragma unroll
            for (int g = 0; g < 3; ++g) {
                tg[g] += __shfl_xor(tg[g], 1, 32);
                tg[g] += __shfl_xor(tg[g], 2, 32);
            }
            float x[3];
#pragma unroll
            for (int g = 0; g < 3; ++g)
                x[g] = (4 * g + q < OO) ? (la[g] - 0.5f * tg[g]) : -1e30f;

            float M = fmaxf(fmaxf(x[0], x[1]), x[2]);
            M = fmaxf(M, __shfl_xor(M, 4, 32));   // across q lanes
            M = fmaxf(M, __shfl_xor(M, 8, 32));

            float e[3], s = 0.0f;
#pragma unroll
            for (int g = 0; g < 3; ++g) { e[g] = expf(x[g] - M); s += e[g]; }
            s += __shfl_xor(s, 4, 32);
            s += __shfl_xor(s, 8, 32);
            const float inv_s = 1.0f / s;
#pragma unroll
            for (int g = 0; g < 3; ++g) rrv[slot][g] = e[g] * inv_s;
        }

        if (k == 0) {
#pragma unroll
            for (int slot = 0; slot < 2; ++slot) {
                const int n = (s0 + 2 * hi + slot) * CC + c;
#pragma unroll
                for (int g = 0; g < 3; ++g) {
                    const int o = 4 * g + q;
                    if (o < OO)
                        rr[((size_t)b * NN + n) * OO + o] = rrv[slot][g];
                }
            }
        }
    }
}

__global__ void zero_kernel(float* __restrict__ p, int n)
{
    const int i = blockIdx.x * blockDim.x + threadIdx.x;
    if (i < n) p[i] = 0.0f;
}

// ---------------------------------------------------------------------------
extern "C" void kernel_launch(void* const* d_in, const int* in_sizes, int n_in,
                              void* d_out, int out_size, void* d_ws, size_t ws_size,
                              hipStream_t stream)
{
    const float* pose   = (const float*)d_in[0];   // (B,H,W,C,16)
    const float* act    = (const float*)d_in[1];   // (B,H,W,C,1)
    const float* w      = (const float*)d_in[2];   // (C,O,4,4)
    const float* beta_a = (const float*)d_in[3];   // (O)
    const float* beta_u = (const float*)d_in[4];   // (O)

    float* out      = (float*)d_out;
    float* out_pose = out;                 // (B,O,16)
    float* out_act  = out + BB * OO * 16;  // (B,O)

    float* ws     = (float*)d_ws;
    float* S0     = ws;                     // B*O
    float* S1     = S0 + BB * OO;           // B*O*16
    float* S2     = S1 + BB * OO * 16;      // B*O*16
    float* mu     = S2 + BB * OO * 16;      // B*O*16
    float* var_   = mu + BB * OO * 16;      // B*O*16
    float* logact = var_ + BB * OO * 16;    // B*O
    float* rr     = logact + BB * OO;       // B*N*O  (16 MB)

    const dim3 mg(7, BB);
    const dim3 mb(224);
    const int  zn = BB * OO * (1 + 16 + 16);  // S0+S1+S2 contiguous

    for (int it = 0; it < 3; ++it) {
        zero_kernel<<<(zn + 255) / 256, 256, 0, stream>>>(S0, zn);
        if (it == 0)
            moments_kernel<1><<<mg, mb, 0, stream>>>(pose, act, w, rr, S0, S1, S2);
        else
            moments_kernel<0><<<mg, mb, 0, stream>>>(pose, act, w, rr, S0, S1, S2);

        const float inv_temp = 1.0f + (float)it;   // 1, 2, 3
        const int   fin      = (it == 2);
        stats_kernel<<<BB, 160, 0, stream>>>(S0, S1, S2, beta_a, beta_u,
                                             mu, var_, logact, inv_temp, fin,
                                             out_pose, out_act);
        if (it < 2)
            rrupd_kernel<<<mg, mb, 0, stream>>>(pose, w, mu, var_, logact, rr);
    }
}